// ExperimentModel_42734924595311
// MI455X (gfx1250) — compile-verified
//
#include <hip/hip_runtime.h>

#define D_MODEL 128
#define D_STATE 64
#define BATCH   32
#define SEQ     4096
#define NCHUNK  32
#define LN_EPS  1e-5f

typedef float v2f __attribute__((ext_vector_type(2)));
typedef float v8f __attribute__((ext_vector_type(8)));

// ---------------------------------------------------------------------------
// Kernel 1: closed-form suffix-sum S4D weights.
//   Wt[t*128+m] = D[m] + sum_n C[m,n]*(1 - exp(g_mn*(L-t)))/(1 - exp(g_mn))
//   g_mn = -exp(log_dt[m]) * exp(A_log[n])
// One block per m; coef/g staged in LDS; 64 v_exp_f32 per (m,t).
// ---------------------------------------------------------------------------
__global__ __launch_bounds__(256) void k1_weights(const float* __restrict__ A_log,
                                                  const float* __restrict__ Dv,
                                                  const float* __restrict__ C_re,
                                                  const float* __restrict__ log_dt,
                                                  float* __restrict__ Wt) {
  const int m = blockIdx.x;
  __shared__ float coef[D_STATE];
  __shared__ float gg[D_STATE];
  __shared__ float baseSh;
  const int tid = threadIdx.x;
  if (tid < D_STATE) {
    float g  = -__expf(log_dt[m]) * __expf(A_log[tid]);
    float cf = C_re[m * D_STATE + tid] / (1.0f - __expf(g));
    coef[tid] = cf;
    gg[tid]   = g;
  }
  __syncthreads();
  if (tid == 0) {
    float s = Dv[m];
    for (int n = 0; n < D_STATE; ++n) s += coef[n];
    baseSh = s;
  }
  __syncthreads();
  const float base = baseSh;
  for (int t = tid; t < SEQ; t += 256) {
    const float f = (float)(SEQ - t);
    float s = 0.0f;
#pragma unroll 8
    for (int n = 0; n < D_STATE; ++n)
      s += coef[n] * __expf(gg[n] * f);     // underflows cleanly to 0 for large |g|*f
    Wt[t * D_MODEL + m] = base - s;
  }
}

// ---------------------------------------------------------------------------
// Kernel 2: fused embedding-gather + LayerNorm + weighted mean-pool partials.
// One wave32 per (b,t) row: each lane owns 4 channels (float4), wave shuffles
// for mean/var, multiply by Wt[t,:], accumulate over this wave's 16 t's,
// deterministic LDS tree-reduce across the 8 waves of the block.
// part[b][chunk][m]  (no atomics -> bitwise deterministic)
// ---------------------------------------------------------------------------
__global__ __launch_bounds__(256) void k2_pool(const int* __restrict__ x,
                                               const float* __restrict__ emb,
                                               const float* __restrict__ ln_w,
                                               const float* __restrict__ ln_b,
                                               const float* __restrict__ Wt,
                                               float* __restrict__ part) {
  const int b     = blockIdx.y;
  const int chunk = blockIdx.x;
  const int lane  = threadIdx.x & 31;
  const int wave  = threadIdx.x >> 5;
  const int m0    = lane * 4;
  const int tbase = (chunk * 8 + wave) * 16;

  const float4 lw = *(const float4*)(ln_w + m0);
  const float4 lb = *(const float4*)(ln_b + m0);

  float4 acc = make_float4(0.f, 0.f, 0.f, 0.f);
  for (int i = 0; i < 16; ++i) {
    const int t   = tbase + i;
    const int tok = x[b * SEQ + t];
    const float4 h = *(const float4*)(emb + (long)tok * D_MODEL + m0);

    float s = h.x + h.y + h.z + h.w;
#pragma unroll
    for (int off = 16; off >= 1; off >>= 1) s += __shfl_xor(s, off, 32);
    const float mu = s * (1.0f / D_MODEL);

    const float dx = h.x - mu, dy = h.y - mu, dz = h.z - mu, dw = h.w - mu;
    float q = dx * dx + dy * dy + dz * dz + dw * dw;
#pragma unroll
    for (int off = 16; off >= 1; off >>= 1) q += __shfl_xor(q, off, 32);
    const float rs = rsqrtf(q * (1.0f / D_MODEL) + LN_EPS);

    const float4 w4 = *(const float4*)(Wt + t * D_MODEL + m0);
    acc.x += (dx * rs * lw.x + lb.x) * w4.x;
    acc.y += (dy * rs * lw.y + lb.y) * w4.y;
    acc.z += (dz * rs * lw.z + lb.z) * w4.z;
    acc.w += (dw * rs * lw.w + lb.w) * w4.w;
  }

  __shared__ float red[8][D_MODEL];
  *(float4*)(&red[wave][m0]) = acc;
  __syncthreads();
  if (threadIdx.x < D_MODEL) {
    float s = 0.f;
#pragma unroll
    for (int w = 0; w < 8; ++w) s += red[w][threadIdx.x];
    part[(b * NCHUNK + chunk) * D_MODEL + threadIdx.x] = s;
  }
}

// ---------------------------------------------------------------------------
// Kernel 3: reduce partials -> pooled[32,128], then classifier GEMM
// out = pooled @ W_cls^T + b_cls via V_WMMA_F32_16X16X4_F32 (fp32 WMMA),
// 2 batch tiles x 32 K-steps, N padded 2 -> 16.
// ---------------------------------------------------------------------------
__global__ __launch_bounds__(256) void k3_final(const float* __restrict__ part,
                                                const float* __restrict__ W_cls,
                                                const float* __restrict__ b_cls,
                                                float* __restrict__ out) {
  __shared__ float pooled[BATCH * D_MODEL];  // [b][m]
  __shared__ float Bpad[D_MODEL * 16];       // [k][n], n padded to 16
  const int tid = threadIdx.x;

  for (int idx = tid; idx < BATCH * D_MODEL; idx += 256) {
    const int b = idx >> 7, m = idx & 127;
    float s = 0.f;
#pragma unroll 8
    for (int c = 0; c < NCHUNK; ++c) s += part[(b * NCHUNK + c) * D_MODEL + m];
    pooled[idx] = s * (1.0f / SEQ);
  }
  for (int idx = tid; idx < D_MODEL * 16; idx += 256) {
    const int k = idx >> 4, n = idx & 15;
    Bpad[idx] = (n < 2) ? W_cls[n * D_MODEL + k] : 0.0f;
  }
  __syncthreads();

  if (tid < 32) {                 // wave 0, EXEC all-1s for WMMA
    const int lane = tid;
    const int half = lane >> 4;   // lanes 0-15: K pair {0,1}; 16-31: {2,3}
    const int lm   = lane & 15;
    for (int bt = 0; bt < 2; ++bt) {
      v8f c = {0.f, 0.f, 0.f, 0.f, 0.f, 0.f, 0.f, 0.f};
      for (int ks = 0; ks < 32; ++ks) {
        const int k = ks * 4 + half * 2;
        v2f a, bm;
        a.x  = pooled[(bt * 16 + lm) * D_MODEL + k];       // A: 16 b-rows x 4 k
        a.y  = pooled[(bt * 16 + lm) * D_MODEL + k + 1];
        bm.x = Bpad[k * 16 + lm];                          // B: 4 k x 16 n
        bm.y = Bpad[(k + 1) * 16 + lm];
        c = __builtin_amdgcn_wmma_f32_16x16x4_f32(false, a, false, bm,
                                                  (short)0, c, false, false);
      }
      if (lm < 2) {               // D layout: VGPR r -> row M=r (+8 for lanes>=16)
#pragma unroll
        for (int r = 0; r < 8; ++r) {
          const int brow = bt * 16 + half * 8 + r;
          out[brow * 2 + lm] = c[r] + b_cls[lm];
        }
      }
    }
  }
}

// ---------------------------------------------------------------------------
extern "C" void kernel_launch(void* const* d_in, const int* in_sizes, int n_in,
                              void* d_out, int out_size, void* d_ws, size_t ws_size,
                              hipStream_t stream) {
  const int*   x      = (const int*)  d_in[0];
  const float* emb    = (const float*)d_in[1];
  const float* ln_w   = (const float*)d_in[2];
  const float* ln_b   = (const float*)d_in[3];
  const float* A_log  = (const float*)d_in[4];
  const float* Dv     = (const float*)d_in[5];
  const float* C_re   = (const float*)d_in[6];
  const float* log_dt = (const float*)d_in[7];
  const float* W_cls  = (const float*)d_in[8];
  const float* b_cls  = (const float*)d_in[9];
  float* out = (float*)d_out;

  float* Wt   = (float*)d_ws;                 // 128*4096 floats = 2 MiB
  float* part = Wt + D_MODEL * SEQ;           // 32*32*128 floats = 512 KiB

  k1_weights<<<dim3(D_MODEL), dim3(256), 0, stream>>>(A_log, Dv, C_re, log_dt, Wt);
  k2_pool  <<<dim3(NCHUNK, BATCH), dim3(256), 0, stream>>>(x, emb, ln_w, ln_b, Wt, part);
  k3_final <<<dim3(1), dim3(256), 0, stream>>>(part, W_cls, b_cls, out);
}